// PointNet2Encoder_11613591568939
// MI455X (gfx1250) — compile-verified
//
#include <hip/hip_runtime.h>

typedef __attribute__((ext_vector_type(2))) float v2f;
typedef __attribute__((ext_vector_type(8))) float v8f;

#define BATCH   8
#define NPTS    4096
#define CIN     9
#define NSAMPLE 32

// ---------------------------------------------------------------------------
// (8,9,4096) -> pts (8,4096,9) and coords (8,4096,3)
// ---------------------------------------------------------------------------
__global__ void transpose_kernel(const float* __restrict__ xyz,
                                 float* __restrict__ pts,
                                 float* __restrict__ coords) {
  int i = blockIdx.x * blockDim.x + threadIdx.x;  // over B*N
  if (i >= BATCH * NPTS) return;
  int b = i / NPTS, p = i - b * NPTS;
  const float* src = xyz + (size_t)b * CIN * NPTS + p;
  float* dp = pts + (size_t)i * CIN;
#pragma unroll
  for (int c = 0; c < CIN; ++c) dp[c] = src[(size_t)c * NPTS];
  coords[i * 3 + 0] = dp[0];
  coords[i * 3 + 1] = dp[1];
  coords[i * 3 + 2] = dp[2];
}

// ---------------------------------------------------------------------------
// Weight prepack: w (Dout x Cin, row-major) ->
//   pk[kk][col] = float2{ w[col][2kk], w[col][2kk+1] },  kk in [0, Kp16/2)
// zero-padded for k >= Cin (Kp16 = K padded to a multiple of 16).
// Exact B-fragment order of V_WMMA_F32_16X16X4_F32: per K-step a lane does
// ONE aligned b64 load; lanes 0-15 of a wave read 128B contiguous.
// ---------------------------------------------------------------------------
__global__ void prepack_w_kernel(const float* __restrict__ w, float* __restrict__ pk,
                                 int Cin, int Dout, int total /* (Kp16/2)*Dout */) {
  int i = blockIdx.x * blockDim.x + threadIdx.x;
  if (i >= total) return;
  int kk = i / Dout, col = i - kk * Dout;
  int k = 2 * kk;
  float x0 = (k < Cin) ? w[(size_t)col * Cin + k] : 0.0f;
  float x1 = (k + 1 < Cin) ? w[(size_t)col * Cin + k + 1] : 0.0f;
  pk[(size_t)i * 2 + 0] = x0;
  pk[(size_t)i * 2 + 1] = x1;
}

// ---------------------------------------------------------------------------
// Farthest point sampling: one workgroup per batch, coords cached in LDS.
// ---------------------------------------------------------------------------
__global__ void fps_kernel(const float* __restrict__ coords, int n, int S,
                           int* __restrict__ fpsidx) {
  extern __shared__ float lds[];
  float* sc = lds;                 // 3*n floats
  float* rv = lds + 3 * n;         // blockDim floats
  int*   ri = (int*)(rv + blockDim.x);
  int b = blockIdx.x, tid = threadIdx.x, nt = blockDim.x;
  for (int i = tid; i < 3 * n; i += nt) sc[i] = coords[(size_t)b * 3 * n + i];
  __syncthreads();

  float dmin[16];
#pragma unroll
  for (int j = 0; j < 16; ++j) dmin[j] = 1e10f;

  int far = 0;
  for (int it = 0; it < S; ++it) {
    if (tid == 0) fpsidx[b * S + it] = far;
    float cx = sc[far * 3 + 0], cy = sc[far * 3 + 1], cz = sc[far * 3 + 2];
    float best = -1.0f;
    int bidx = 0, j = 0;
    for (int p = tid; p < n; p += nt, ++j) {
      float dx = sc[p * 3 + 0] - cx, dy = sc[p * 3 + 1] - cy, dz = sc[p * 3 + 2] - cz;
      float d = dx * dx + dy * dy + dz * dz;
      float dm = dmin[j];
      dm = (d < dm) ? d : dm;
      dmin[j] = dm;
      if (dm > best) { best = dm; bidx = p; }
    }
    rv[tid] = best; ri[tid] = bidx;
    __syncthreads();
    for (int off = nt >> 1; off > 0; off >>= 1) {
      if (tid < off) {
        float v2 = rv[tid + off]; int i2 = ri[tid + off];
        if (v2 > rv[tid] || (v2 == rv[tid] && i2 < ri[tid])) { rv[tid] = v2; ri[tid] = i2; }
      }
      __syncthreads();
    }
    far = ri[0];
    __syncthreads();
  }
}

// ---------------------------------------------------------------------------
// new_xyz[b,s,:] = coords[b, fps_idx[b,s], :]
// ---------------------------------------------------------------------------
__global__ void gather_newxyz_kernel(const float* __restrict__ coords,
                                     const int* __restrict__ fpsidx,
                                     float* __restrict__ newxyz,
                                     int n, int S, int total) {
  int i = blockIdx.x * blockDim.x + threadIdx.x;  // over B*S
  if (i >= total) return;
  int b = i / S;
  int p = fpsidx[i];
  size_t src = ((size_t)b * n + p) * 3;
  newxyz[i * 3 + 0] = coords[src + 0];
  newxyz[i * 3 + 1] = coords[src + 1];
  newxyz[i * 3 + 2] = coords[src + 2];
}

// ---------------------------------------------------------------------------
// Ball query: one wave32 per query, ballot-append in ascending index order.
// ---------------------------------------------------------------------------
__global__ void ballquery_kernel(const float* __restrict__ coords,
                                 const float* __restrict__ newxyz,
                                 int* __restrict__ ballidx,
                                 int n, int S, float r2, int nq) {
  __shared__ int sel[8][NSAMPLE];
  int wave = threadIdx.x >> 5, lane = threadIdx.x & 31;
  int q = blockIdx.x * 8 + wave;
  int cnt = 0;
  if (q < nq) {
    int b = q / S;
    float cx = newxyz[q * 3 + 0], cy = newxyz[q * 3 + 1], cz = newxyz[q * 3 + 2];
    const float* cb = coords + (size_t)b * n * 3;
    for (int j0 = 0; j0 < n && cnt < NSAMPLE; j0 += 32) {
      int j = j0 + lane;
      bool inr = false;
      if (j < n) {
        float dx = cb[j * 3 + 0] - cx, dy = cb[j * 3 + 1] - cy, dz = cb[j * 3 + 2] - cz;
        inr = (dx * dx + dy * dy + dz * dz <= r2);
      }
      unsigned mask = (unsigned)__ballot(inr);
      int pos = cnt + __popc(mask & ((1u << lane) - 1u));
      if (inr && pos < NSAMPLE) sel[wave][pos] = j;
      cnt += __popc(mask);
    }
  }
  __syncthreads();
  if (q < nq) {
    int c = cnt < NSAMPLE ? cnt : NSAMPLE;
    int first = (c > 0) ? sel[wave][0] : (n - 1);
    ballidx[q * NSAMPLE + lane] = (lane < c) ? sel[wave][lane] : first;
  }
}

// ---------------------------------------------------------------------------
// One MLP sublayer via V_WMMA_F32_16X16X4_F32, prepacked zero-padded weights.
// Kp is a multiple of 16 -> steps is a multiple of 4 -> every group of 4
// WMMA-steps is full: straight-line 4x ds_load_b64 + 4x global_load_b64,
// one wait, 4 WMMAs, ping-pong double buffered, no lane or element guards.
// Tail prefetch base is clamped (scalar cselect) so addresses stay valid.
// ---------------------------------------------------------------------------
__device__ __forceinline__ void mlp_sublayer_pk(
    const float* __restrict__ in, int inStride, int Kp,
    const float* __restrict__ pkw, const float* __restrict__ bias,
    const float* __restrict__ gamma, const float* __restrict__ beta,
    float* __restrict__ outb, int Dout, int outStride,
    int lane, int wave, int nWaves) {
  int nTiles = 2 * (Dout >> 4);            // 2 M-tiles x Dout/16 N-tiles
  int hi = lane >> 4, lo = lane & 15;
  int steps = Kp >> 2;                     // multiple of 4, >= 4
  size_t bD = (size_t)Dout;                // B row stride in v2f units
  for (int t = wave; t < nTiles; t += nWaves) {
    int mt = t & 1, ntile = t >> 1;
    int col = ntile * 16 + lo;
    const float* arow = in + (mt * 16 + lo) * inStride;       // LDS, 8B aligned
    const v2f* bbase = (const v2f*)pkw + col + (size_t)hi * bD;
    v8f acc = {};

    auto loadG = [&](v2f* av, v2f* bv, int nb) {
      const float* ap = arow + nb * 4 + 2 * hi;
      const v2f* bp = bbase + (size_t)(nb * 2) * bD;
#pragma unroll
      for (int j = 0; j < 4; ++j) {
        av[j] = *(const v2f*)(ap + 4 * j);                    // ds_load_b64
        bv[j] = bp[(size_t)(2 * j) * bD];                     // global_load_b64
      }
    };
    auto compG = [&](const v2f* av, const v2f* bv) {
#pragma unroll
      for (int j = 0; j < 4; ++j)
        acc = __builtin_amdgcn_wmma_f32_16x16x4_f32(
            false, av[j], false, bv[j], (short)0, acc, false, false);
    };

    v2f a0[4], b0[4], a1[4], b1[4];
    loadG(a0, b0, 0);
    int base = 0;
    while (true) {
      int nb = base + 4; nb = nb < steps ? nb : steps - 4;    // scalar clamp
      loadG(a1, b1, nb);          // prefetch next group (always valid addrs)
      compG(a0, b0);              // compute current group
      base += 4;
      if (base >= steps) break;
      nb = base + 4; nb = nb < steps ? nb : steps - 4;
      loadG(a0, b0, nb);
      compG(a1, b1);
      base += 4;
      if (base >= steps) break;
    }

    float gg = gamma[col], bb = bias[col], ee = beta[col];
#pragma unroll
    for (int i = 0; i < 8; ++i) {
      int row = mt * 16 + hi * 8 + i;                         // C/D layout
      float v = gg * (acc[i] + bb) + ee;
      outb[row * outStride + col] = v > 0.0f ? v : 0.0f;
    }
  }
}

// ---------------------------------------------------------------------------
// Fused grouping + 3-layer MLP + maxpool. One workgroup per (b,s) group.
// LDS row strides are = 4 (mod 8) floats to spread b64 A-loads over 64 banks.
// ---------------------------------------------------------------------------
__global__ void __launch_bounds__(256) group_mlp_kernel(
    const float* __restrict__ coords, const float* __restrict__ feats,
    const float* __restrict__ newxyz, const int* __restrict__ ballidx,
    const float* pk0, const float* b0, const float* g0, const float* e0,
    const float* pk1, const float* b1, const float* g1, const float* e1,
    const float* pk2, const float* b2, const float* g2, const float* e2,
    float* __restrict__ out,
    int n, int S, int Cf, int C0p, int D0, int D1, int D2,
    int strideA, int strideB) {
  extern __shared__ float lds[];
  float* bufA = lds;
  float* bufB = lds + 32 * strideA;
  int g = blockIdx.x;
  int b = g / S;
  int tid = threadIdx.x;
  int lane = tid & 31, wave = tid >> 5, nWaves = blockDim.x >> 5;
  int C0 = Cf + 3;
  const int* bi = ballidx + g * NSAMPLE;
  float ncx = newxyz[g * 3 + 0], ncy = newxyz[g * 3 + 1], ncz = newxyz[g * 3 + 2];

  // Stage grouped input (32 x C0p), zero-padded out to C0p (multiple of 16).
  for (int i = tid; i < NSAMPLE * C0p; i += blockDim.x) {
    int k = i / C0p, c = i - k * C0p;
    int p = bi[k];
    float v = 0.0f;
    if (c == 0)      v = coords[((size_t)b * n + p) * 3 + 0] - ncx;
    else if (c == 1) v = coords[((size_t)b * n + p) * 3 + 1] - ncy;
    else if (c == 2) v = coords[((size_t)b * n + p) * 3 + 2] - ncz;
    else if (c < C0) v = feats[((size_t)b * n + p) * Cf + (c - 3)];
    bufA[k * strideA + c] = v;
  }
  __syncthreads();
  mlp_sublayer_pk(bufA, strideA, C0p, pk0, b0, g0, e0, bufB, D0, strideB, lane, wave, nWaves);
  __syncthreads();
  mlp_sublayer_pk(bufB, strideB, D0, pk1, b1, g1, e1, bufA, D1, strideA, lane, wave, nWaves);
  __syncthreads();
  mlp_sublayer_pk(bufA, strideA, D1, pk2, b2, g2, e2, bufB, D2, strideB, lane, wave, nWaves);
  __syncthreads();
  // Max-pool over the 32 group points.
  for (int o = tid; o < D2; o += blockDim.x) {
    float m = bufB[o];
    for (int k = 1; k < NSAMPLE; ++k) m = fmaxf(m, bufB[k * strideB + o]);
    out[(size_t)g * D2 + o] = m;
  }
}

// ---------------------------------------------------------------------------
// Head: mean over S points, then two FC layers.
// ---------------------------------------------------------------------------
__global__ void mean_kernel(const float* __restrict__ in, float* __restrict__ out,
                            int S, int C, int total) {
  int i = blockIdx.x * blockDim.x + threadIdx.x;
  if (i >= total) return;
  int b = i / C, c = i - b * C;
  float s = 0.0f;
  for (int k = 0; k < S; ++k) s += in[((size_t)b * S + k) * C + c];
  out[i] = s / (float)S;
}

__global__ void fc_kernel(const float* __restrict__ in, const float* __restrict__ w,
                          const float* __restrict__ bias, float* __restrict__ out,
                          int Cin, int Cout, int total, int relu) {
  int i = blockIdx.x * blockDim.x + threadIdx.x;
  if (i >= total) return;
  int b = i / Cout, o = i - b * Cout;
  const float* ir = in + (size_t)b * Cin;
  const float* wr = w + (size_t)o * Cin;
  float s = 0.0f;
  for (int k = 0; k < Cin; ++k) s += ir[k] * wr[k];
  s += bias[o];
  if (relu) s = s > 0.0f ? s : 0.0f;
  out[i] = s;
}

// ---------------------------------------------------------------------------
// Host orchestration
// ---------------------------------------------------------------------------
extern "C" void kernel_launch(void* const* d_in, const int* in_sizes, int n_in,
                              void* d_out, int out_size, void* d_ws, size_t ws_size,
                              hipStream_t stream) {
  (void)in_sizes; (void)n_in; (void)out_size; (void)ws_size;
  const float* xyz = (const float*)d_in[0];
  // params flat order: 4 SA layers x 3 sublayers x (w,bias,gamma,beta), then w1,b1,w2,b2
  const float* P[52];
  for (int i = 0; i < 52; ++i) P[i] = (const float*)d_in[1 + i];

  char* base = (char*)d_ws;
  size_t off = 0;
  auto alloc = [&](size_t bytes) -> char* {
    char* p = base + off;
    off += bytes;
    off = (off + 255) & ~(size_t)255;
    return p;
  };

  float* pts     = (float*)alloc((size_t)BATCH * NPTS * CIN * 4);
  float* coords0 = (float*)alloc((size_t)BATCH * 4096 * 3 * 4);
  float* coords1 = (float*)alloc((size_t)BATCH * 1024 * 3 * 4);
  float* coords2 = (float*)alloc((size_t)BATCH * 256 * 3 * 4);
  float* coords3 = (float*)alloc((size_t)BATCH * 64 * 3 * 4);
  float* coords4 = (float*)alloc((size_t)BATCH * 16 * 3 * 4);
  float* feat1   = (float*)alloc((size_t)BATCH * 1024 * 64 * 4);
  float* feat2   = (float*)alloc((size_t)BATCH * 256 * 128 * 4);
  float* feat3   = (float*)alloc((size_t)BATCH * 64 * 256 * 4);
  float* feat4   = (float*)alloc((size_t)BATCH * 16 * 512 * 4);
  int*   fpsbuf  = (int*)alloc((size_t)BATCH * 1024 * 4);
  int*   ballbuf = (int*)alloc((size_t)BATCH * 1024 * NSAMPLE * 4);
  float* meanbuf = (float*)alloc((size_t)BATCH * 512 * 4);
  float* hbuf    = (float*)alloc((size_t)BATCH * 512 * 4);

  // Layer configs
  int   nIn[4]  = {4096, 1024, 256, 64};
  int   Sg[4]   = {1024, 256, 64, 16};
  int   Cf[4]   = {9, 64, 128, 256};
  int   C0p[4]  = {16, 80, 144, 272};     // pad16(Cf+3): full WMMA groups
  int   D0[4]   = {32, 64, 128, 256};
  int   D1[4]   = {32, 64, 128, 256};
  int   D2[4]   = {64, 128, 256, 512};
  float r2[4]   = {0.01f, 0.04f, 0.16f, 0.64f};

  // Prepacked weights: per sublayer (Cin, Kp16, Dout)
  float* pkw[12];
  int pkCin[12], pkKp[12], pkDout[12];
  for (int l = 0; l < 4; ++l) {
    pkCin[l * 3 + 0] = Cf[l] + 3; pkKp[l * 3 + 0] = C0p[l]; pkDout[l * 3 + 0] = D0[l];
    pkCin[l * 3 + 1] = D0[l];     pkKp[l * 3 + 1] = D0[l];  pkDout[l * 3 + 1] = D1[l];
    pkCin[l * 3 + 2] = D1[l];     pkKp[l * 3 + 2] = D1[l];  pkDout[l * 3 + 2] = D2[l];
  }
  for (int s = 0; s < 12; ++s)
    pkw[s] = (float*)alloc((size_t)pkKp[s] * pkDout[s] * 4);

  // 0) prepack all 12 weight matrices (tiny; L2-resident thereafter)
  for (int s = 0; s < 12; ++s) {
    int total = (pkKp[s] / 2) * pkDout[s];
    const float* w = P[(s / 3) * 12 + (s % 3) * 4];  // layer*12 + sub*4 + 0
    prepack_w_kernel<<<(total + 255) / 256, 256, 0, stream>>>(
        w, pkw[s], pkCin[s], pkDout[s], total);
  }

  // 1) transpose
  {
    int total = BATCH * NPTS;
    transpose_kernel<<<(total + 255) / 256, 256, 0, stream>>>(xyz, pts, coords0);
  }

  // 2) SA layers
  float* coordsIn[4]  = {coords0, coords1, coords2, coords3};
  float* coordsOut[4] = {coords1, coords2, coords3, coords4};
  const float* featIn[4] = {pts, feat1, feat2, feat3};
  float* featOut[4]      = {feat1, feat2, feat3, feat4};

  for (int l = 0; l < 4; ++l) {
    int n = nIn[l], S = Sg[l];
    // FPS: one workgroup per batch
    size_t fpsLds = (size_t)3 * n * 4 + 256 * 8;
    fps_kernel<<<BATCH, 256, fpsLds, stream>>>(coordsIn[l], n, S, fpsbuf);
    // gather centroids
    int nq = BATCH * S;
    gather_newxyz_kernel<<<(nq + 255) / 256, 256, 0, stream>>>(
        coordsIn[l], fpsbuf, coordsOut[l], n, S, nq);
    // ball query: 8 wave32 queries per block
    ballquery_kernel<<<(nq + 7) / 8, 256, 0, stream>>>(
        coordsIn[l], coordsOut[l], ballbuf, n, S, r2[l], nq);
    // fused group + MLP x3 + maxpool
    int maxA = C0p[l] > D1[l] ? C0p[l] : D1[l];
    int maxB = D0[l] > D2[l] ? D0[l] : D2[l];
    // pad so stride = 4 (mod 8) floats -> b64 A-loads spread over all banks
    int sA = maxA + ((maxA % 8) == 4 ? 8 : 4);
    int sB = maxB + ((maxB % 8) == 4 ? 8 : 4);
    size_t mlpLds = (size_t)32 * (sA + sB) * 4;
    int pb = l * 12;  // param base (3 sublayers x 4)
    int ps = l * 3;   // prepacked base
    group_mlp_kernel<<<nq, 256, mlpLds, stream>>>(
        coordsIn[l], featIn[l], coordsOut[l], ballbuf,
        pkw[ps + 0], P[pb + 1], P[pb + 2], P[pb + 3],
        pkw[ps + 1], P[pb + 5], P[pb + 6], P[pb + 7],
        pkw[ps + 2], P[pb + 9], P[pb + 10], P[pb + 11],
        featOut[l], n, S, Cf[l], C0p[l], D0[l], D1[l], D2[l], sA, sB);
  }

  // 3) head
  {
    int total = BATCH * 512;
    mean_kernel<<<(total + 255) / 256, 256, 0, stream>>>(feat4, meanbuf, 16, 512, total);
    fc_kernel<<<(total + 255) / 256, 256, 0, stream>>>(
        meanbuf, P[48], P[49], hbuf, 512, 512, total, 1);
    fc_kernel<<<(total + 255) / 256, 256, 0, stream>>>(
        hbuf, P[50], P[51], (float*)d_out, 512, 512, total, 0);
  }
}